// ProposalLayer_34763465294052
// MI455X (gfx1250) — compile-verified
//
#include <hip/hip_runtime.h>

// ---------------- problem constants ----------------
#define NUM_B   16
#define A_NUM   9
#define HWCELLS 14400            // 120*120
#define N_ANCH  129600           // 9 * 14400
#define PRE     3000
#define POST    300
#define ARRN    4096             // bitonic size (>= PRE, pow2)
#define NWORD   47               // ceil(3000/64)
#define NMS_THR 0.6f
#define CLIP_HI 1919.0f

// workspace layout (bytes)
#define WS_TOPK_OFF  0                         // 16*3000*4      = 192,000
#define WS_BOX_OFF   192000                    // 16*3000*16     = 768,000
#define WS_MASK_OFF  960000                    // 16*3000*47*8   = 18,048,000

__device__ __forceinline__ unsigned sortable_u32(float f) {
  unsigned u = __float_as_uint(f);
  return (u & 0x80000000u) ? ~u : (u | 0x80000000u);
}

// -------------------------------------------------------------------------
// K1: per-image exact top-3000 (value desc, index asc) via 8-bit radix
//     select + compaction + 4096-wide bitonic sort in LDS.
// scores[b, p] = cls[(b*N_ANCH + q)*2]  where q=(a,hw), p = hw*9 + a
// -------------------------------------------------------------------------
__global__ __launch_bounds__(1024) void topk_kernel(const float* __restrict__ cls,
                                                    int* __restrict__ topkIdx) {
  const int b = blockIdx.x;
  const int t = threadIdx.x;
  const float* base = cls + (size_t)b * N_ANCH * 2;

  __shared__ unsigned hist[256];
  __shared__ unsigned sPrefix, sK;
  __shared__ unsigned cntAbove, cntEq;
  __shared__ unsigned long long arr[ARRN];   // 32KB

  if (t == 0) { sPrefix = 0u; sK = PRE; }
  __syncthreads();

  for (int r = 0; r < 4; ++r) {
    const int shift = 24 - 8 * r;
    if (t < 256) hist[t] = 0u;
    __syncthreads();
    const unsigned maskHi = (r == 0) ? 0u : (0xFFFFFFFFu << (32 - 8 * r));
    const unsigned pref   = sPrefix;
    for (int q = t; q < N_ANCH; q += 1024) {
      unsigned u = sortable_u32(base[2 * q]);
      if ((u & maskHi) == (pref & maskHi))
        atomicAdd(&hist[(u >> shift) & 255u], 1u);
    }
    __syncthreads();
    if (t == 0) {
      unsigned K = sK, cum = 0u;
      for (int bin = 255; bin >= 0; --bin) {
        unsigned c = hist[bin];
        if (cum + c >= K) { sK = K - cum; sPrefix = pref | ((unsigned)bin << shift); break; }
        cum += c;
      }
    }
    __syncthreads();
  }

  const unsigned T      = sPrefix;     // exact sortable key of rank-3000 element
  const unsigned needEq = sK;          // #keys == T to keep
  const unsigned nAbove = PRE - needEq;
  if (t == 0) { cntAbove = 0u; cntEq = 0u; }
  __syncthreads();

  for (int q = t; q < N_ANCH; q += 1024) {
    unsigned u = sortable_u32(base[2 * q]);
    if (u < T) continue;
    int a = q / HWCELLS, hw = q - a * HWCELLS;
    unsigned p = (unsigned)hw * 9u + (unsigned)a;
    unsigned long long pk = ((unsigned long long)u << 32) | (unsigned)(~p);
    if (u > T) {
      unsigned s = atomicAdd(&cntAbove, 1u);
      arr[s] = pk;
    } else {
      unsigned e = atomicAdd(&cntEq, 1u);
      if (e < needEq) arr[nAbove + e] = pk;
    }
  }
  for (int s = PRE + t; s < ARRN; s += 1024) arr[s] = 0ull;
  __syncthreads();

  // bitonic sort, descending on packed (key, ~idx)
  for (int k = 2; k <= ARRN; k <<= 1) {
    for (int j = k >> 1; j > 0; j >>= 1) {
      for (int i = t; i < ARRN; i += 1024) {
        int ixj = i ^ j;
        if (ixj > i) {
          unsigned long long va = arr[i], vb = arr[ixj];
          bool up = ((i & k) == 0);
          if (up ? (va < vb) : (va > vb)) { arr[i] = vb; arr[ixj] = va; }
        }
      }
      __syncthreads();
    }
  }

  for (int r = t; r < PRE; r += 1024)
    topkIdx[b * PRE + r] = (int)(~((unsigned)arr[r]));
}

// -------------------------------------------------------------------------
// K2: decode + clip boxes for the 16*3000 selected anchors.
// anchor geometry: ctr = 16*cell + 8; (w,h) from ratio {23/12,16/16,11/22} * scale {8,16,32}
// -------------------------------------------------------------------------
__global__ __launch_bounds__(256) void decode_kernel(const float* __restrict__ bbox,
                                                     const int* __restrict__ topkIdx,
                                                     float4* __restrict__ boxes) {
  int g = blockIdx.x * 256 + threadIdx.x;
  if (g >= NUM_B * PRE) return;
  int b = g / PRE;
  int p = topkIdx[g];
  int a = p % 9, hw = p / 9;
  int wx = hw % 120, hy = hw / 120;
  int rr = a / 3, ss = a % 3;
  float scale = (float)(8 << ss);
  float wbase = (rr == 0) ? 23.f : ((rr == 1) ? 16.f : 11.f);
  float hbase = (rr == 0) ? 12.f : ((rr == 1) ? 16.f : 22.f);
  float W = wbase * scale, H = hbase * scale;

  const float4* dp = (const float4*)bbox + ((size_t)(b * A_NUM + a) * HWCELLS + hw);
  float4 d = *dp;

  float cx = (float)(wx * 16 + 8) + d.x * W;
  float cy = (float)(hy * 16 + 8) + d.y * H;
  float pw = expf(d.z) * W;
  float ph = expf(d.w) * H;
  float4 o;
  o.x = fminf(fmaxf(cx - 0.5f * pw, 0.f), CLIP_HI);
  o.y = fminf(fmaxf(cy - 0.5f * ph, 0.f), CLIP_HI);
  o.z = fminf(fmaxf(cx + 0.5f * pw, 0.f), CLIP_HI);
  o.w = fminf(fmaxf(cy + 0.5f * ph, 0.f), CLIP_HI);
  boxes[g] = o;
}

// -------------------------------------------------------------------------
// K3: suppression bitmask matrix. One 64-bit word per (i, jword) pair,
//     bit set for j>i with IoU>0.6. Boxes staged to LDS with CDNA5 async
//     global->LDS copies (ASYNCcnt-tracked).
// -------------------------------------------------------------------------
__global__ __launch_bounds__(256) void mask_kernel(const float4* __restrict__ boxes,
                                                   unsigned long long* __restrict__ maskArr) {
  const int b = blockIdx.y;
  __shared__ float4 sbox[PRE];     // 48 KB
  __shared__ float  sarea[PRE];    // 12 KB

  // async stage: each lane copies one 16B box global->LDS (gfx1250 async path)
  {
    const float4* gsrc = boxes + (size_t)b * PRE;
    for (int i = threadIdx.x; i < PRE; i += 256) {
      unsigned lds_off = (unsigned)(unsigned long long)(&sbox[i]);
      const float4* gp = gsrc + i;
      asm volatile("global_load_async_to_lds_b128 %0, %1, off"
                   :: "v"(lds_off), "v"(gp) : "memory");
    }
    asm volatile("s_wait_asynccnt 0" ::: "memory");
  }
  __syncthreads();
  for (int i = threadIdx.x; i < PRE; i += 256) {
    float4 v = sbox[i];
    sarea[i] = (v.z - v.x + 1.f) * (v.w - v.y + 1.f);
  }
  __syncthreads();

  int t = blockIdx.x * 256 + threadIdx.x;
  if (t >= PRE * NWORD) return;
  int i = t / NWORD, wq = t - i * NWORD;
  float4 bi = sbox[i];
  float  ai = sarea[i];
  unsigned long long word = 0ull;
  int j0 = wq * 64;
  int jend = (j0 + 64 < PRE) ? (j0 + 64) : PRE;
  int jstart = (j0 > i + 1) ? j0 : (i + 1);
  for (int j = jstart; j < jend; ++j) {
    float4 bj = sbox[j];
    float iw = fminf(bi.z, bj.z) - fmaxf(bi.x, bj.x) + 1.f;
    float ih = fminf(bi.w, bj.w) - fmaxf(bi.y, bj.y) + 1.f;
    iw = fmaxf(iw, 0.f); ih = fmaxf(ih, 0.f);
    float inter = iw * ih;
    float iou = inter / (ai + sarea[j] - inter);
    if (iou > NMS_THR) word |= (1ull << (j - j0));
  }
  maskArr[((size_t)b * PRE + i) * NWORD + wq] = word;
}

// -------------------------------------------------------------------------
// K4: serial NMS scan (one wave32 per image; supp state in registers,
//     alive test scalarized via v_readlane; rows pipelined 8 deep with
//     global_prefetch ahead) + compaction of first 300 survivors.
// -------------------------------------------------------------------------
__global__ __launch_bounds__(32) void nms_scan_kernel(const unsigned long long* __restrict__ maskArr,
                                                      const float4* __restrict__ boxes,
                                                      float* __restrict__ out) {
  const int b = blockIdx.x;
  const int l = threadIdx.x;                       // wave32 lane
  const unsigned long long* M = maskArr + (size_t)b * PRE * NWORD;
  const bool hi_ok = (l + 32) < NWORD;             // lanes 0..14 own words 32..46

  unsigned long long w0 = 0ull, w1 = 0ull;         // suppression state
  unsigned long long r0[8], r1[8];                 // pipelined row buffers
#pragma unroll
  for (int d = 0; d < 8; ++d) {
    r0[d] = M[(size_t)d * NWORD + l];
    r1[d] = hi_ok ? M[(size_t)d * NWORD + l + 32] : 0ull;
  }

  for (int ii = 0; ii < PRE; ii += 8) {
#pragma unroll
    for (int d = 0; d < 8; ++d) {
      const int i = ii + d;
      const int wi = i >> 6;
      unsigned long long sel = (wi < 32) ? w0 : w1;
      int owner = wi & 31;
      unsigned lo = (unsigned)__builtin_amdgcn_readlane((int)(unsigned)sel, owner);
      unsigned hi = (unsigned)__builtin_amdgcn_readlane((int)(unsigned)(sel >> 32), owner);
      unsigned long long sval = ((unsigned long long)hi << 32) | lo;
      bool alive = ((sval >> (i & 63)) & 1ull) == 0ull;
      if (alive) { w0 |= r0[d]; w1 |= r1[d]; }
      int nx = i + 8;
      if (nx < PRE) {
        r0[d] = M[(size_t)nx * NWORD + l];
        r1[d] = hi_ok ? M[(size_t)nx * NWORD + l + 32] : 0ull;
      }
    }
    if (ii + 64 < PRE)
      __builtin_prefetch(&M[(size_t)(ii + 64) * NWORD + l], 0, 1);
  }

  // ---- emit: kept boxes in top-k order, first 300 rows; rest zeros ----
  float* ob = out + (size_t)b * POST * 5;
  int basecnt = 0;
  for (int g = 0; g < (PRE + 31) / 32 && basecnt < POST; ++g) {
    int idx = g * 32 + l;
    int wi = g >> 1;
    unsigned long long sel = (wi < 32) ? w0 : w1;
    int owner = wi & 31;
    unsigned lo = (unsigned)__builtin_amdgcn_readlane((int)(unsigned)sel, owner);
    unsigned hi = (unsigned)__builtin_amdgcn_readlane((int)(unsigned)(sel >> 32), owner);
    unsigned long long word = ((unsigned long long)hi << 32) | lo;
    int bit = (g & 1) * 32 + l;
    bool keep = (idx < PRE) && (((word >> bit) & 1ull) == 0ull);
    unsigned mlow = (unsigned)__ballot(keep);
    int rank = basecnt + __popc(mlow & ((1u << l) - 1u));
    if (keep && rank < POST) {
      float4 v = boxes[(size_t)b * PRE + idx];
      float* row = ob + (size_t)rank * 5;
      row[0] = (float)b; row[1] = v.x; row[2] = v.y; row[3] = v.z; row[4] = v.w;
    }
    basecnt += __popc(mlow);
  }
  int kept = (basecnt < POST) ? basecnt : POST;
  for (int r = kept + l; r < POST; r += 32) {
    float* row = ob + (size_t)r * 5;
    row[0] = (float)b; row[1] = 0.f; row[2] = 0.f; row[3] = 0.f; row[4] = 0.f;
  }
}

// -------------------------------------------------------------------------
extern "C" void kernel_launch(void* const* d_in, const int* in_sizes, int n_in,
                              void* d_out, int out_size, void* d_ws, size_t ws_size,
                              hipStream_t stream) {
  const float* cls  = (const float*)d_in[0];   // (16,9,120,120,2) f32
  const float* bbox = (const float*)d_in[1];   // (16,9,120,120,4) f32
  float* out = (float*)d_out;                  // (16,300,5) f32
  char* ws = (char*)d_ws;

  int*                topkIdx = (int*)(ws + WS_TOPK_OFF);
  float4*             boxes   = (float4*)(ws + WS_BOX_OFF);
  unsigned long long* maskArr = (unsigned long long*)(ws + WS_MASK_OFF);

  topk_kernel<<<NUM_B, 1024, 0, stream>>>(cls, topkIdx);

  decode_kernel<<<(NUM_B * PRE + 255) / 256, 256, 0, stream>>>(bbox, topkIdx, boxes);

  dim3 gmask((PRE * NWORD + 255) / 256, NUM_B);
  mask_kernel<<<gmask, 256, 0, stream>>>(boxes, maskArr);

  nms_scan_kernel<<<NUM_B, 32, 0, stream>>>(maskArr, boxes, out);
}